// ParametrisedUnPooling_67070209294883
// MI455X (gfx1250) — compile-verified
//
#include <hip/hip_runtime.h>

// Fused ParametrisedUnPooling for MI455X (gfx1250, wave32).
// out[i, :] = sum_k w[r,k] * x_in[idx[r,k], :]   where r = indices_target[i]
//
// - One wave32 == one output row (32 lanes x 2 float4 == 1KB row).
// - Wave-uniform metadata forced into SGPRs (readfirstlane) -> scalar branch
//   select + saddr-form global_load_b128 gathers.
// - x_in (51MB) is L2-resident (192MB L2, ~20x row reuse): output stores are
//   NON-TEMPORAL (TH=NT) so the 205MB write stream does not evict x_in.
// - Each wave handles 2 rows: row i1's neighbor rows are staged into LDS with
//   gfx1250 async global->LDS loads (ASYNCcnt) while row i0 is gathered
//   directly; row i1 is then consumed from LDS. Fallback: global_prefetch_b8.

#define DEVFN __device__ __forceinline__

typedef float v4f __attribute__((ext_vector_type(4)));
typedef int   v4i __attribute__((vector_size(16)));        // matches builtin proto
typedef __attribute__((address_space(1))) v4i* g_v4i_ptr;  // global int4*
typedef __attribute__((address_space(3))) v4i* l_v4i_ptr;  // LDS int4*

#if defined(__has_builtin)
#if __has_builtin(__builtin_amdgcn_global_load_async_to_lds_b128)
#define HAVE_ASYNC_LDS 1
#endif
#endif
#ifndef HAVE_ASYNC_LDS
#define HAVE_ASYNC_LDS 0
#endif

DEVFN int rfl(int v) { return __builtin_amdgcn_readfirstlane(v); }

#if HAVE_ASYNC_LDS
// Async DMA: 16 bytes per lane, global -> LDS.  Tracked by ASYNCcnt.
DEVFN void async_gather16(const float* g, float* l)
{
    // AS1 VA == flat VA for global memory (int round-trip also drops const).
    // LDS needs a real addrspacecast (flat aperture VA != LDS offset).
    __builtin_amdgcn_global_load_async_to_lds_b128(
        (g_v4i_ptr)(uintptr_t)g,
        (l_v4i_ptr)l,
        0, 0);
}

DEVFN void wait_async0()
{
#if __has_builtin(__builtin_amdgcn_s_wait_asynccnt)
    __builtin_amdgcn_s_wait_asynccnt(0);
#else
    asm volatile("s_wait_asynccnt 0x0" ::: "memory");
#endif
    asm volatile("" ::: "memory");  // keep LDS reads below the wait
}
#endif

// ---- Row compute, source = global gather (saddr-form b128 loads) ----
template <int DEG>
DEVFN void do_row_g(const float* __restrict__ x_in,
                    const int*   __restrict__ ip,
                    const float* __restrict__ wp,
                    float*       __restrict__ op,
                    int lane)
{
    int   s[DEG];
    float w[DEG];
#pragma unroll
    for (int k = 0; k < DEG; ++k) {
        s[k] = rfl(ip[k]);
        w[k] = wp[k];
    }

    v4f a[DEG], b[DEG];
#pragma unroll
    for (int k = 0; k < DEG; ++k) {      // all 2*DEG b128 loads in flight
        const v4f* src = (const v4f*)(x_in + (size_t)s[k] * 256);
        a[k] = src[lane];
        b[k] = src[lane + 32];
    }

    v4f accA = {0.f, 0.f, 0.f, 0.f};
    v4f accB = {0.f, 0.f, 0.f, 0.f};
#pragma unroll
    for (int k = 0; k < DEG; ++k) {
        accA += w[k] * a[k];
        accB += w[k] * b[k];
    }
    __builtin_nontemporal_store(accA, (v4f*)op + lane);        // TH=NT: don't
    __builtin_nontemporal_store(accB, (v4f*)op + lane + 32);   // evict x_in
}

#if HAVE_ASYNC_LDS
// ---- Row compute, source = LDS staging buffer (ds_load_b128) ----
template <int DEG>
DEVFN void do_row_l(const float* __restrict__ slice,
                    const float* __restrict__ wp,
                    float*       __restrict__ op,
                    int lane)
{
    float w[DEG];
#pragma unroll
    for (int k = 0; k < DEG; ++k) w[k] = wp[k];

    v4f accA = {0.f, 0.f, 0.f, 0.f};
    v4f accB = {0.f, 0.f, 0.f, 0.f};
#pragma unroll
    for (int k = 0; k < DEG; ++k) {
        const v4f* src = (const v4f*)(slice + k * 256);
        v4f a = src[lane];
        v4f b = src[lane + 32];
        accA += w[k] * a;
        accB += w[k] * b;
    }
    __builtin_nontemporal_store(accA, (v4f*)op + lane);
    __builtin_nontemporal_store(accB, (v4f*)op + lane + 32);
}
#endif

DEVFN void resolve(int r, int n4, int n6,
                   const int* __restrict__ i4, const int* __restrict__ i6,
                   const int* __restrict__ i7,
                   const float* __restrict__ w4, const float* __restrict__ w6,
                   const float* __restrict__ w7,
                   const int** ip, const float** wp, int* deg)
{
    if (r < n4) {
        *ip = i4 + (size_t)r * 4;  *wp = w4 + (size_t)r * 4;  *deg = 4;
    } else if (r < n4 + n6) {
        int t = r - n4;
        *ip = i6 + (size_t)t * 6;  *wp = w6 + (size_t)t * 6;  *deg = 6;
    } else {
        int t = r - n4 - n6;
        *ip = i7 + (size_t)t * 7;  *wp = w7 + (size_t)t * 7;  *deg = 7;
    }
}

DEVFN void dispatch_g(int deg, const float* x_in,
                      const int* ip, const float* wp, float* op, int lane)
{
    switch (deg) {
        case 4:  do_row_g<4>(x_in, ip, wp, op, lane); break;
        case 6:  do_row_g<6>(x_in, ip, wp, op, lane); break;
        default: do_row_g<7>(x_in, ip, wp, op, lane); break;
    }
}

#if HAVE_ASYNC_LDS
DEVFN void dispatch_l(int deg, const float* slice,
                      const float* wp, float* op, int lane)
{
    switch (deg) {
        case 4:  do_row_l<4>(slice, wp, op, lane); break;
        case 6:  do_row_l<6>(slice, wp, op, lane); break;
        default: do_row_l<7>(slice, wp, op, lane); break;
    }
}
#endif

__global__ __launch_bounds__(256)
void unpool_fused_kernel(const float* __restrict__ x_in,
                         const int*   __restrict__ i4,
                         const int*   __restrict__ i6,
                         const int*   __restrict__ i7,
                         const int*   __restrict__ tgt,
                         const float* __restrict__ w4,
                         const float* __restrict__ w6,
                         const float* __restrict__ w7,
                         float*       __restrict__ out,
                         int n4, int n6, int n_out)
{
#if HAVE_ASYNC_LDS
    __shared__ float stage[8 * 7 * 256];       // 7KB per wave, 56KB per block
#endif
    const int lane = threadIdx.x & 31;
    const int wv   = threadIdx.x >> 5;
    const int gw   = rfl(blockIdx.x * 8 + wv); // global wave id (SGPR)
    const int i0   = gw * 2;
    const int i1   = i0 + 1;

    // --- Resolve row i1 metadata and start moving its gather data on-chip.
    const int*   ip1  = nullptr;
    const float* wp1  = nullptr;
    int          deg1 = 0;
    const bool   has1 = (i1 < n_out);
    if (has1) {
        int r1 = rfl(tgt[i1]);
        resolve(r1, n4, n6, i4, i6, i7, w4, w6, w7, &ip1, &wp1, &deg1);
#if HAVE_ASYNC_LDS
        float* slice = &stage[wv * (7 * 256)];
#pragma unroll
        for (int k = 0; k < 7; ++k) {
            if (k < deg1) {
                int s = rfl(ip1[k]);
                const float* g = x_in + (size_t)s * 256 + (lane << 2);
                float*       l = slice + k * 256 + (lane << 2);
                async_gather16(g, l);              // 512B per instruction
                async_gather16(g + 128, l + 128);  // second half of the row
            }
        }
#else
#pragma unroll
        for (int k = 0; k < 7; ++k) {
            if (k < deg1) {
                int s = rfl(ip1[k]);
                __builtin_prefetch(x_in + (size_t)s * 256 + lane * 8, 0, 3);
            }
        }
#endif
    }

    // --- Row i0: direct gather + weighted sum + fused-permutation NT store.
    if (i0 < n_out) {
        int r0 = rfl(tgt[i0]);
        const int*   ip0;
        const float* wp0;
        int          deg0;
        resolve(r0, n4, n6, i4, i6, i7, w4, w6, w7, &ip0, &wp0, &deg0);
        dispatch_g(deg0, x_in, ip0, wp0, out + (size_t)i0 * 256, lane);
    }

    // --- Row i1: consume the staged data.
    if (has1) {
#if HAVE_ASYNC_LDS
        wait_async0();
        dispatch_l(deg1, &stage[wv * (7 * 256)], wp1,
                   out + (size_t)i1 * 256, lane);
#else
        dispatch_g(deg1, x_in, ip1, wp1, out + (size_t)i1 * 256, lane);
#endif
    }
}

extern "C" void kernel_launch(void* const* d_in, const int* in_sizes, int n_in,
                              void* d_out, int out_size, void* d_ws, size_t ws_size,
                              hipStream_t stream)
{
    const float* x_in = (const float*)d_in[0];
    const int*   i4   = (const int*)  d_in[1];
    const int*   i6   = (const int*)  d_in[2];
    const int*   i7   = (const int*)  d_in[3];
    const int*   tgt  = (const int*)  d_in[4];
    const float* w4   = (const float*)d_in[5];
    const float* w6   = (const float*)d_in[6];
    const float* w7   = (const float*)d_in[7];
    float*       out  = (float*)d_out;

    const int n4    = in_sizes[1] / 4;
    const int n6    = in_sizes[2] / 6;
    const int n_out = in_sizes[4];

    const int waves  = (n_out + 1) / 2;      // 2 output rows per wave32
    const int blocks = (waves + 7) / 8;      // 8 waves per 256-thread block

    unpool_fused_kernel<<<dim3(blocks), dim3(256), 0, stream>>>(
        x_in, i4, i6, i7, tgt, w4, w6, w7, out, n4, n6, n_out);
}